// TopNGenerator_36206574305938
// MI455X (gfx1250) — compile-verified
//
#include <hip/hip_runtime.h>
#include <math.h>

#define BB   512
#define DD   512
#define NCOS 64
#define HID  32
#define MAXN 8192
#define CC   256
#define NSEL 1024

typedef float v2f __attribute__((ext_vector_type(2)));
typedef float v4f __attribute__((ext_vector_type(4)));
typedef float v8f __attribute__((ext_vector_type(8)));

// ---------------- Kernel 1: angle MLP + L2 normalize -> ang [B, NCOS] -------
__global__ __launch_bounds__(256) void mlp_ang_kernel(
    const float* __restrict__ latent, const float* __restrict__ w1,
    const float* __restrict__ b1, const float* __restrict__ w2,
    const float* __restrict__ b2, float* __restrict__ ang_out)
{
    int r = blockIdx.x * 256 + threadIdx.x;
    if (r >= BB) return;

    float h[HID];
    #pragma unroll
    for (int j = 0; j < HID; ++j) h[j] = b1[j];
    const float* lrow = latent + (size_t)r * DD;
    for (int k = 0; k < DD; ++k) {
        float lv = lrow[k];
        const float* wr = w1 + (size_t)k * HID;
        #pragma unroll
        for (int j = 0; j < HID; ++j) h[j] = fmaf(lv, wr[j], h[j]);
    }
    #pragma unroll
    for (int j = 0; j < HID; ++j) h[j] = fmaxf(h[j], 0.0f);

    float a[NCOS];
    #pragma unroll
    for (int j = 0; j < NCOS; ++j) a[j] = b2[j];
    for (int k = 0; k < HID; ++k) {
        float hv = h[k];
        const float* wr = w2 + (size_t)k * NCOS;
        #pragma unroll
        for (int j = 0; j < NCOS; ++j) a[j] = fmaf(hv, wr[j], a[j]);
    }
    float ss = 0.0f;
    #pragma unroll
    for (int j = 0; j < NCOS; ++j) ss = fmaf(a[j], a[j], ss);
    float scale = 1.0f / (sqrtf(ss) + 1e-5f);
    float* orow = ang_out + (size_t)r * NCOS;
    #pragma unroll
    for (int j = 0; j < NCOS; ++j) orow[j] = a[j] * scale;
}

// ---------------- Kernel 2: scores = ang @ anchors^T via WMMA f32 16x16x4 ---
// grid: (MAXN/128, BB/16), block 256 (8 waves); wave w does 16x16 tile at
// (m0 = by*16, j0 = bx*128 + w*16), K-loop of 16 wmma steps (K=64).
__global__ __launch_bounds__(256) void score_wmma_kernel(
    const float* __restrict__ ang, const float* __restrict__ anchors,
    float* __restrict__ scores)
{
    const int lane = threadIdx.x & 31;
    const int wave = threadIdx.x >> 5;
    const int half = lane >> 4;   // 0: K pair {0,1}, 1: K pair {2,3}
    const int r    = lane & 15;   // row (A: M offset) / col (B: N offset)
    const int m0 = blockIdx.y * 16;
    const int j0 = blockIdx.x * 128 + wave * 16;

    const float* aRow = ang     + (size_t)(m0 + r) * NCOS + 2 * half;
    const float* bRow = anchors + (size_t)(j0 + r) * NCOS + 2 * half;

    v8f acc = {};
    #pragma unroll
    for (int kb = 0; kb < 16; ++kb) {
        v2f av = *(const v2f*)(aRow + kb * 4);
        v2f bv = *(const v2f*)(bRow + kb * 4);
        acc = __builtin_amdgcn_wmma_f32_16x16x4_f32(
            /*neg_a=*/false, av, /*neg_b=*/false, bv,
            /*c_mod=*/(short)0, acc, /*reuse_a=*/false, /*reuse_b=*/false);
    }
    // C/D layout: VGPR v -> M = v + 8*half, N = r
    float* out0 = scores + (size_t)(m0 + half * 8) * MAXN + j0 + r;
    #pragma unroll
    for (int v = 0; v < 8; ++v) out0[(size_t)v * MAXN] = acc[v];
}

// ---------------- Kernel 3: softmax stats + bitonic top-1024 + gather ------
__global__ __launch_bounds__(256) void topk_out_kernel(
    const float* __restrict__ scores, const float* __restrict__ points,
    const float* __restrict__ l1w, const float* __restrict__ l1b,
    const float* __restrict__ l2w, const float* __restrict__ l2b,
    float* __restrict__ out)
{
    __shared__ unsigned long long keys[MAXN];   // exactly 64 KB LDS
    const int tid = threadIdx.x;
    const int row = blockIdx.x;

    // Keep this row's 32 scores per thread in registers (b128 loads).
    // Thread owns elements i = t*1024 + tid*4 + q, t in 0..7, q in 0..3.
    float sc[32];
    const v4f* srow4 = (const v4f*)(scores + (size_t)row * MAXN);
    #pragma unroll
    for (int t = 0; t < 8; ++t) {
        v4f v = srow4[t * 256 + tid];
        sc[4 * t + 0] = v.x; sc[4 * t + 1] = v.y;
        sc[4 * t + 2] = v.z; sc[4 * t + 3] = v.w;
    }

    float* fs = (float*)keys;   // scratch overlay (before keys are filled)

    // ---- row max ----
    float lmax = sc[0];
    #pragma unroll
    for (int t = 1; t < 32; ++t) lmax = fmaxf(lmax, sc[t]);
    for (int o = 16; o > 0; o >>= 1) lmax = fmaxf(lmax, __shfl_xor(lmax, o, 32));
    if ((tid & 31) == 0) fs[tid >> 5] = lmax;
    __syncthreads();
    if (tid == 0) {
        float m = fs[0];
        for (int w = 1; w < 8; ++w) m = fmaxf(m, fs[w]);
        fs[8] = m;
    }
    __syncthreads();
    const float rowmax = fs[8];

    // ---- row sum of exp(score - max) ----
    float lsum = 0.0f;
    #pragma unroll
    for (int t = 0; t < 32; ++t) lsum += expf(sc[t] - rowmax);
    for (int o = 16; o > 0; o >>= 1) lsum += __shfl_xor(lsum, o, 32);
    if ((tid & 31) == 0) fs[16 + (tid >> 5)] = lsum;
    __syncthreads();
    if (tid == 0) {
        float s = fs[16];
        for (int w = 1; w < 8; ++w) s += fs[16 + w];
        fs[9] = s;
    }
    __syncthreads();
    const float rowsum = fs[9];
    __syncthreads();   // everyone has read scratch before keys overwrite it

    // ---- build sort keys: ascending sort => descending score, ties by index
    #pragma unroll
    for (int t = 0; t < 8; ++t) {
        #pragma unroll
        for (int q = 0; q < 4; ++q) {
            int i = t * 1024 + tid * 4 + q;
            unsigned u = __float_as_uint(sc[4 * t + q]);
            u = (u & 0x80000000u) ? ~u : (u | 0x80000000u);   // order-preserving
            unsigned inv = ~u;                                // largest -> smallest
            keys[i] = ((unsigned long long)inv << 32) | (unsigned)i;
        }
    }
    __syncthreads();

    // ---- bitonic sort (ascending) over 8192 u64 keys ----
    for (unsigned k = 2; k <= MAXN; k <<= 1) {
        for (unsigned j = k >> 1; j > 0; j >>= 1) {
            for (unsigned i = tid; i < MAXN; i += 256) {
                unsigned ixj = i ^ j;
                if (ixj > i) {
                    unsigned long long a = keys[i];
                    unsigned long long b = keys[ixj];
                    bool up = ((i & k) == 0);
                    if (up == (a > b)) { keys[i] = b; keys[ixj] = a; }
                }
            }
            __syncthreads();
        }
    }

    // ---- s[e] = n * softmax(score) for top-1024 (overlay past entry 2048) --
    float* sbuf = (float*)(keys + 2048);
    for (int e = tid; e < NSEL; e += 256) {
        unsigned inv = (unsigned)(keys[e] >> 32);
        unsigned u = ~inv;
        unsigned fb = (u & 0x80000000u) ? (u ^ 0x80000000u) : ~u;
        float scv = __uint_as_float(fb);
        sbuf[e] = (float)NSEL * (expf(scv - rowmax) / rowsum);
    }
    __syncthreads();

    // ---- gather + affine, B128: thread = 4 channels of one of 4 points ----
    // Output is write-once -> non-temporal (keep points/scores L2-resident).
    const int grp = tid >> 6;        // which of 4 selected points this iter
    const int cq  = tid & 63;        // float4 column (channels 4*cq..4*cq+3)
    const v4f w1v = ((const v4f*)l1w)[cq];
    const v4f b1v = ((const v4f*)l1b)[cq];
    const v4f w2v = ((const v4f*)l2w)[cq];
    const v4f b2v = ((const v4f*)l2b)[cq];
    const v4f* pts4 = (const v4f*)points;
    v4f* out4 = (v4f*)(out + ((size_t)row * NSEL) * CC);
    for (int e = 0; e < NSEL; e += 4) {
        int ee = e + grp;
        unsigned idx = (unsigned)(keys[ee] & 0xFFFFFFFFu);
        float s = sbuf[ee];
        v4f pv = pts4[(size_t)idx * (CC / 4) + cq];
        v4f rv;
        rv.x = fmaf(s, w1v.x, b1v.x) * pv.x + fmaf(s, w2v.x, b2v.x);
        rv.y = fmaf(s, w1v.y, b1v.y) * pv.y + fmaf(s, w2v.y, b2v.y);
        rv.z = fmaf(s, w1v.z, b1v.z) * pv.z + fmaf(s, w2v.z, b2v.z);
        rv.w = fmaf(s, w1v.w, b1v.w) * pv.w + fmaf(s, w2v.w, b2v.w);
        __builtin_nontemporal_store(rv, out4 + (size_t)ee * (CC / 4) + cq);
    }
}

// ---------------------------------------------------------------------------
extern "C" void kernel_launch(void* const* d_in, const int* in_sizes, int n_in,
                              void* d_out, int out_size, void* d_ws, size_t ws_size,
                              hipStream_t stream)
{
    (void)in_sizes; (void)n_in; (void)out_size; (void)ws_size;
    const float* latent  = (const float*)d_in[0];
    // d_in[1] is n (==1024, structural; compile-time NSEL)
    const float* points  = (const float*)d_in[2];
    const float* anchors = (const float*)d_in[3];
    const float* w1      = (const float*)d_in[4];
    const float* b1      = (const float*)d_in[5];
    const float* w2      = (const float*)d_in[6];
    const float* b2      = (const float*)d_in[7];
    const float* l1w     = (const float*)d_in[8];
    const float* l1b     = (const float*)d_in[9];
    const float* l2w     = (const float*)d_in[10];
    const float* l2b     = (const float*)d_in[11];
    float* out = (float*)d_out;

    float* ang    = (float*)d_ws;          // BB*NCOS floats (128 KB)
    float* scores = ang + BB * NCOS;       // BB*MAXN floats (16 MB)

    mlp_ang_kernel<<<dim3(BB / 256), 256, 0, stream>>>(latent, w1, b1, w2, b2, ang);
    score_wmma_kernel<<<dim3(MAXN / 128, BB / 16), 256, 0, stream>>>(ang, anchors, scores);
    topk_out_kernel<<<dim3(BB), 256, 0, stream>>>(scores, points, l1w, l1b, l2w, l2b, out);
}